// CustomLSTM_49538152792349
// MI455X (gfx1250) — compile-verified
//
#include <hip/hip_runtime.h>

// ---------------------------------------------------------------------------
// Scalar LSTM (input=1, hidden=1) over T steps: a pure serial recurrence.
// Latency-bound: the only lever is the per-step dependency chain.
//   - All four gate nonlinearities via gfx1250 hardware v_tanh_f32
//     (sigmoid(x) = 0.5*tanh(x/2)+0.5, the 0.5 folded into weights/biases).
//   - x streamed wave-wide: one coalesced 128B load per 32 steps, register
//     double-buffered, + global_prefetch_b8 ahead; v_readlane broadcasts.
//   - h/c pinned to VGPRs via opaque v_mov so the serial chain stays in
//     VALU (no SALU<->VALU ping-pong around v_tanh, which is VALU-only).
//     The x-side terms remain uniform/SALU and co-execute off the chain.
// ---------------------------------------------------------------------------

__device__ __forceinline__ float fast_tanh(float x) {
#if __has_builtin(__builtin_amdgcn_tanhf)
  return __builtin_amdgcn_tanhf(x);           // v_tanh_f32 (gfx1250 TRANS op)
#else
  // tanh(x) = sign(x) * (1 - 2*e/(1+e)),  e = 2^(-2*log2(e)*|x|)
  float ax = __builtin_fabsf(x);
  float e  = __builtin_amdgcn_exp2f(-2.8853900817779268f * ax);
  float r  = __builtin_amdgcn_rcpf(1.0f + e);
  float t  = __builtin_fmaf(-2.0f * e, r, 1.0f);
  return __builtin_copysignf(t, x);
#endif
}

struct LstmParams {
  // Gates i,f,o pre-scaled by 0.5 so sigmoid(g) = 0.5*tanh(g') + 0.5
  // with g' = h*wh + fma(x, wi, bb). Gate g (cell input) unscaled.
  float wi0, wh0, bb0;   // i
  float wi1, wh1, bb1;   // f
  float wi2, wh2, bb2;   // g
  float wi3, wh3, bb3;   // o
};

__device__ __forceinline__ void lstm_step(float xt, float& h, float& c,
                                          const LstmParams& p) {
  // x-only terms: wave-uniform, independent of h -> SALU float ops that the
  // scheduler hoists ahead of (and co-executes with) the VALU chain.
  float ai = __builtin_fmaf(xt, p.wi0, p.bb0);
  float af = __builtin_fmaf(xt, p.wi1, p.bb1);
  float ag = __builtin_fmaf(xt, p.wi2, p.bb2);
  float ao = __builtin_fmaf(xt, p.wi3, p.bb3);
  // Chain from h (all VALU, h/c are VGPR-divergent by construction):
  // v_fma -> v_tanh -> v_fma per gate, gates in parallel.
  float ti = fast_tanh(__builtin_fmaf(h, p.wh0, ai));
  float tf = fast_tanh(__builtin_fmaf(h, p.wh1, af));
  float gg = fast_tanh(__builtin_fmaf(h, p.wh2, ag));
  float to = fast_tanh(__builtin_fmaf(h, p.wh3, ao));
  float i = __builtin_fmaf(ti, 0.5f, 0.5f);
  float f = __builtin_fmaf(tf, 0.5f, 0.5f);
  float o = __builtin_fmaf(to, 0.5f, 0.5f);
  c = __builtin_fmaf(f, c, i * gg);
  h = o * fast_tanh(c);
}

__device__ __forceinline__ float readlane_f(float v, int lane) {
  int i = __builtin_amdgcn_readlane(__builtin_bit_cast(int, v), lane);
  return __builtin_bit_cast(float, i);
}

__global__ __launch_bounds__(32, 1) void
lstm_seq_kernel(const float* __restrict__ x,
                const float* __restrict__ w_ih,
                const float* __restrict__ w_hh,
                const float* __restrict__ b_ih,
                const float* __restrict__ b_hh,
                float* __restrict__ out, int n) {
  LstmParams p;
  p.wi0 = 0.5f * w_ih[0]; p.wh0 = 0.5f * w_hh[0]; p.bb0 = 0.5f * (b_ih[0] + b_hh[0]);
  p.wi1 = 0.5f * w_ih[1]; p.wh1 = 0.5f * w_hh[1]; p.bb1 = 0.5f * (b_ih[1] + b_hh[1]);
  p.wi2 =        w_ih[2]; p.wh2 =        w_hh[2]; p.bb2 =        (b_ih[2] + b_hh[2]);
  p.wi3 = 0.5f * w_ih[3]; p.wh3 = 0.5f * w_hh[3]; p.bb3 = 0.5f * (b_ih[3] + b_hh[3]);

  const int lane    = threadIdx.x;           // wave32: 0..31
  const int nChunks = n >> 5;                // 32 x-values per chunk

  // Pin the recurrent state to VGPRs via an opaque v_mov: the compiler's
  // uniformity analysis must treat h/c as divergent, keeping the entire
  // serial chain in VALU (v_fma_f32 / v_tanh_f32) with no per-step
  // v_readfirstlane/SALU round trips on the critical path.
  float h, c;
  asm volatile("v_mov_b32 %0, 0" : "=v"(h));
  asm volatile("v_mov_b32 %0, 0" : "=v"(c));

  // Register double-buffer: lane L holds x[chunk*32 + L].
  float xv = (nChunks > 0) ? x[lane] : 0.0f;

  for (int ch = 0; ch < nChunks; ++ch) {
    const float xcur = xv;
    // Issue next chunk's coalesced 128B load before chewing on this chunk.
    if (ch + 1 < nChunks) {
      xv = x[(long)(ch + 1) * 32 + lane];
    }
    // gfx1250 prefetch path (global_prefetch_b8), a few chunks ahead.
    if (ch + 8 < nChunks) {
      __builtin_prefetch(&x[(long)(ch + 8) * 32 + lane], 0, 0);
    }
#pragma unroll
    for (int j = 0; j < 32; ++j) {
      const float xt = readlane_f(xcur, j);  // v_readlane_b32, imm lane
      lstm_step(xt, h, c, p);
    }
  }

  // Generic tail (T = 20M is divisible by 32, so normally skipped).
  for (int t = nChunks << 5; t < n; ++t) {
    lstm_step(x[t], h, c, p);
  }

  if (lane == 0) {
    out[0] = h;
  }
}

extern "C" void kernel_launch(void* const* d_in, const int* in_sizes, int n_in,
                              void* d_out, int out_size, void* d_ws, size_t ws_size,
                              hipStream_t stream) {
  (void)n_in; (void)out_size; (void)d_ws; (void)ws_size;
  const float* x    = (const float*)d_in[0];
  const float* w_ih = (const float*)d_in[1];
  const float* w_hh = (const float*)d_in[2];
  const float* b_ih = (const float*)d_in[3];
  const float* b_hh = (const float*)d_in[4];
  float* out = (float*)d_out;
  const int n = in_sizes[0];

  // Strictly serial recurrence: one wave owns the whole sequence.
  lstm_seq_kernel<<<1, 32, 0, stream>>>(x, w_ih, w_hh, b_ih, b_hh, out, n);
}